// GCN_79955111182893
// MI455X (gfx1250) — compile-verified
//
#include <hip/hip_runtime.h>
#include <hip/hip_bf16.h>

#define N_NODES 50000
#define N_EDGES 800000
#define N_GRAPHS 512
#define DFEAT 128

typedef __attribute__((ext_vector_type(2))) float v2f;
typedef __attribute__((ext_vector_type(8))) float v8f;

// ---------------------------------------------------------------- utilities

__global__ void zero_f32(float* __restrict__ p, int n) {
    int i = blockIdx.x * blockDim.x + threadIdx.x;
    if (i < n) p[i] = 0.0f;
}

__global__ void degree_accum(const int* __restrict__ idx, float* deg, int n) {
    int i = blockIdx.x * blockDim.x + threadIdx.x;
    if (i < n) atomicAdd(&deg[idx[i]], 1.0f);
}

__global__ void inv_sqrt_deg(float* p, int n) {
    int i = blockIdx.x * blockDim.x + threadIdx.x;
    if (i < n) {
        float d = p[i];
        p[i] = (d > 0.0f) ? rsqrtf(fmaxf(d, 1.0f)) : 0.0f;
    }
}

// out[node][f] = in[node][f] * norm[node]   (in == out allowed)
__global__ void scale_rows(const float* in, const float* __restrict__ norm,
                           float* out, int total) {
    int i = blockIdx.x * blockDim.x + threadIdx.x;
    if (i < total) {
        int node = i >> 7; // DFEAT == 128
        out[i] = in[i] * norm[node];
    }
}

// SpMM scatter: agg[dst[e]] += h[src[e]]; 32 threads per edge, float4 each.
__global__ void spmm_scatter(const float* __restrict__ h,
                             const int* __restrict__ src,
                             const int* __restrict__ dst,
                             float* agg, int nedges) {
    long long idx = (long long)blockIdx.x * blockDim.x + threadIdx.x;
    int e = (int)(idx >> 5);
    if (e >= nedges) return;
    int f = ((int)idx & 31) * 4;
    int s = src[e], d = dst[e];
    const float4 v = *(const float4*)(h + (size_t)s * DFEAT + f);
    float* o = agg + (size_t)d * DFEAT + f;
    atomicAdd(o + 0, v.x);
    atomicAdd(o + 1, v.y);
    atomicAdd(o + 2, v.z);
    atomicAdd(o + 3, v.w);
}

// ----------------------------------------------------- WMMA fp32 GEMM + bias
// out[M,128] = act(A[M,128] @ W[128,128] + bias), using V_WMMA_F32_16X16X4_F32.
// One wave per 16-row strip; 4 waves per block; W staged in LDS (64 KB).
//
// A-frag (16x4 f32): lanes 0-15 -> M=lane, K={0,1} in v0/v1;
//                    lanes 16-31 -> M=lane-16, K={2,3} in v0/v1.
// B-frag (4x16 f32): mirrored (N in lane&15, K half selected by lane>>4).
// C/D (16x16 f32):   VGPR i -> M = i + 8*(lane>>4), N = lane&15.
__global__ __launch_bounds__(128) void gcn_gemm_wmma(
    const float* __restrict__ A, const float* __restrict__ W,
    const float* __restrict__ bias, float* __restrict__ out,
    int M, int do_relu) {
    __shared__ float sW[DFEAT * DFEAT]; // 64 KB of the WGP's 320 KB LDS

    const int tid = threadIdx.x;
    {   // cooperative 128-thread float4 copy of W into LDS
        const float4* Wv = (const float4*)W;
        float4* sWv = (float4*)sW;
        for (int i = tid; i < DFEAT * DFEAT / 4; i += 128) sWv[i] = Wv[i];
    }
    __syncthreads();

    const int wave = tid >> 5;
    const int lane = tid & 31;
    const int m0 = (blockIdx.x * 4 + wave) * 16;
    if (m0 >= M) return; // wave-uniform; EXEC stays all-ones for WMMA

    const int mrow  = lane & 15;
    const int khalf = (lane >> 4) << 1; // 0 or 2

    // Register-resident A strip fragments: 32 k-steps x 2 floats = 64 VGPRs
    float a0[32], a1[32];
    const float* Arow = A + (size_t)(m0 + mrow) * DFEAT + khalf;
#pragma unroll
    for (int ks = 0; ks < 32; ++ks) {
        a0[ks] = Arow[ks * 4 + 0];
        a1[ks] = Arow[ks * 4 + 1];
    }

    const int ncol  = lane & 15;
    const int crow0 = m0 + ((lane >> 4) << 3);

#pragma unroll
    for (int nt = 0; nt < 8; ++nt) {
        v8f acc = {0.f, 0.f, 0.f, 0.f, 0.f, 0.f, 0.f, 0.f};
        const float* Wp = sW + khalf * DFEAT + nt * 16 + ncol;
#pragma unroll
        for (int ks = 0; ks < 32; ++ks) {
            v2f a; a.x = a0[ks];            a.y = a1[ks];
            v2f b; b.x = Wp[ks * 4 * DFEAT]; b.y = Wp[ks * 4 * DFEAT + DFEAT];
            // 8-arg form: (neg_a, A, neg_b, B, c_mod, C, reuse_a, reuse_b)
            acc = __builtin_amdgcn_wmma_f32_16x16x4_f32(
                false, a, false, b, (short)0, acc, false, false);
        }
        const float bv = bias[nt * 16 + ncol];
#pragma unroll
        for (int i = 0; i < 8; ++i) {
            float v = acc[i] + bv;
            if (do_relu) v = fmaxf(v, 0.0f);
            out[(size_t)(crow0 + i) * DFEAT + nt * 16 + ncol] = v;
        }
    }
}

// ------------------------------------------------------------ graph pooling

__global__ void pool_feat(const float* __restrict__ h,
                          const int* __restrict__ gids,
                          float* gsums, int total) {
    int i = blockIdx.x * blockDim.x + threadIdx.x;
    if (i < total) {
        int node = i >> 7;
        int f = i & 127;
        atomicAdd(&gsums[(size_t)gids[node] * DFEAT + f], h[i]);
    }
}

__global__ void pool_cnt(const int* __restrict__ gids, float* gcnt, int n) {
    int i = blockIdx.x * blockDim.x + threadIdx.x;
    if (i < n) atomicAdd(&gcnt[gids[i]], 1.0f);
}

// out[g][c] = (gsums[g]/max(cnt,1)) . Wc[:,c] + bc[c];  one block per graph
__global__ __launch_bounds__(128) void classify(
    const float* __restrict__ gsums, const float* __restrict__ gcnt,
    const float* __restrict__ Wc, const float* __restrict__ bc,
    float* __restrict__ out) {
    __shared__ float s0[128], s1[128];
    int g = blockIdx.x;
    int f = threadIdx.x;
    float inv = 1.0f / fmaxf(gcnt[g], 1.0f);
    float hv = gsums[(size_t)g * DFEAT + f] * inv;
    s0[f] = hv * Wc[f * 2 + 0];
    s1[f] = hv * Wc[f * 2 + 1];
    __syncthreads();
    for (int off = 64; off > 0; off >>= 1) {
        if (f < off) { s0[f] += s0[f + off]; s1[f] += s1[f + off]; }
        __syncthreads();
    }
    if (f == 0) {
        out[g * 2 + 0] = s0[0] + bc[0];
        out[g * 2 + 1] = s1[0] + bc[1];
    }
}

// ---------------------------------------------------------------- launcher

extern "C" void kernel_launch(void* const* d_in, const int* in_sizes, int n_in,
                              void* d_out, int out_size, void* d_ws, size_t ws_size,
                              hipStream_t stream) {
    const float* features = (const float*)d_in[0];
    const int*   src      = (const int*)  d_in[1];
    const int*   dst      = (const int*)  d_in[2];
    const int*   gids     = (const int*)  d_in[3];
    const float* W1 = (const float*)d_in[4];
    const float* b1 = (const float*)d_in[5];
    const float* W2 = (const float*)d_in[6];
    const float* b2 = (const float*)d_in[7];
    const float* Wc = (const float*)d_in[8];
    const float* bc = (const float*)d_in[9];
    float* out = (float*)d_out;

    const int N = N_NODES, E = N_EDGES;
    const int NF = N * DFEAT;

    float* ws    = (float*)d_ws;
    float* onorm = ws;                 // [N]  D_out^{-1/2} (src side)
    float* inorm = ws + N;             // [N]  D_in^{-1/2}  (dst side)
    float* bufA  = ws + 2 * N;         // [N,128]
    float* bufB  = bufA + NF;          // [N,128]
    float* gsums = bufB + NF;          // [512,128]
    float* gcnt  = gsums + N_GRAPHS * DFEAT; // [512]

    const int T = 256;
    auto blk = [](int n, int t) { return (n + t - 1) / t; };

    // degrees -> inv sqrt norms
    zero_f32<<<blk(2 * N, T), T, 0, stream>>>(onorm, 2 * N);
    degree_accum<<<blk(E, T), T, 0, stream>>>(src, onorm, E);
    degree_accum<<<blk(E, T), T, 0, stream>>>(dst, inorm, E);
    inv_sqrt_deg<<<blk(2 * N, T), T, 0, stream>>>(onorm, 2 * N);

    // -------- layer 1: relu((D_i^-1/2 * S * (D_o^-1/2 * X)) @ W1 + b1)
    scale_rows<<<blk(NF, T), T, 0, stream>>>(features, onorm, bufA, NF);
    zero_f32<<<blk(NF, T), T, 0, stream>>>(bufB, NF);
    spmm_scatter<<<blk(E * 32, T), T, 0, stream>>>(bufA, src, dst, bufB, E);
    scale_rows<<<blk(NF, T), T, 0, stream>>>(bufB, inorm, bufB, NF);
    gcn_gemm_wmma<<<blk(N / 16, 4), 128, 0, stream>>>(bufB, W1, b1, bufA, N, 1);

    // -------- layer 2 (no relu)
    scale_rows<<<blk(NF, T), T, 0, stream>>>(bufA, onorm, bufB, NF);
    zero_f32<<<blk(NF, T), T, 0, stream>>>(bufA, NF);
    spmm_scatter<<<blk(E * 32, T), T, 0, stream>>>(bufB, src, dst, bufA, E);
    scale_rows<<<blk(NF, T), T, 0, stream>>>(bufA, inorm, bufA, NF);
    gcn_gemm_wmma<<<blk(N / 16, 4), 128, 0, stream>>>(bufA, W2, b2, bufB, N, 0);

    // -------- mean-pool per graph + classifier
    zero_f32<<<blk(N_GRAPHS * DFEAT + N_GRAPHS, T), T, 0, stream>>>(
        gsums, N_GRAPHS * DFEAT + N_GRAPHS);
    pool_feat<<<blk(NF, T), T, 0, stream>>>(bufB, gids, gsums, NF);
    pool_cnt<<<blk(N, T), T, 0, stream>>>(gids, gcnt, N);
    classify<<<N_GRAPHS, 128, 0, stream>>>(gsums, gcnt, Wc, bc, out);
}